// LSTM_NATTEN_60662118088753
// MI455X (gfx1250) — compile-verified
//
#include <hip/hip_runtime.h>
#include <cstdint>
#include <cstddef>

// ---------------------------------------------------------------------------
// Model constants (from reference)
// ---------------------------------------------------------------------------
#define B_   2
#define T_   128          // grid H
#define P_   88           // grid W
#define C_   128
#define NH_  4            // heads
#define DH_  32           // head dim
#define HH_  64           // lstm hidden per direction
#define G4_  256          // 4*HH
#define NSEQ (B_*P_)      // 176
#define NPIX (B_*T_*P_)   // 22528  (= NSEQ*T_)

typedef __attribute__((ext_vector_type(2))) float v2f;
typedef __attribute__((ext_vector_type(8))) float v8f;

// fp32 WMMA: D(16x16) = A(16x4) * B(4x16) + C   (CDNA5 matrix path)
__device__ __forceinline__ v8f wmma4(v2f a, v2f b, v8f c) {
  return __builtin_amdgcn_wmma_f32_16x16x4_f32(
      /*neg_a=*/false, a, /*neg_b=*/false, b,
      /*c_mod=*/(short)0, c, /*reuse_a=*/false, /*reuse_b=*/false);
}

__device__ __forceinline__ float sigm(float x) { return 1.f / (1.f + __expf(-x)); }

__device__ __forceinline__ float waveSum(float v) {
#pragma unroll
  for (int m = 16; m >= 1; m >>= 1) v += __shfl_xor(v, m, 32);
  return v;
}

// ---------------------------------------------------------------------------
// WMMA GEMM with transposed B: out[M,N] = A[M,K] x Bt[N,K]^T + bias (+resid)
// Bt element (k,n) = Bt[n*ldb + k]  -> contiguous v2f fragment loads.
// Each wave computes a 16x64 strip; grid.x*8 waves cover (M/16)*(N/64) units.
// All shapes tile exactly -> no divergence anywhere around WMMA.
// ---------------------------------------------------------------------------
template <bool HASB2, bool HASRES>
__global__ __launch_bounds__(256) void gemm16t(
    const float* __restrict__ A, int lda,
    const float* __restrict__ Bt, int ldb,
    const float* __restrict__ bias1, const float* __restrict__ bias2,
    const float* __restrict__ resid, float* __restrict__ out, int ldo,
    int NG, int K)
{
  const int wave = threadIdx.x >> 5;
  const int lane = threadIdx.x & 31;
  const int unit = blockIdx.x * 8 + wave;
  const int mt = unit / NG;
  const int ng = unit % NG;
  const int m0 = mt * 16;
  const int n0 = ng * 64;
  const int row  = lane & 15;
  const int half = lane >> 4;      // 0: K+0/1, C rows 0..7 ; 1: K+2/3, C rows 8..15
  const int koff = half * 2;

  const float* ap = A + (size_t)(m0 + row) * lda + koff;
  const float* bp0 = Bt + (size_t)(n0 + row) * ldb + koff;
  const size_t btile = (size_t)16 * ldb;

  // init accumulators with the (column-broadcast) bias
  v8f acc[4];
#pragma unroll
  for (int t2 = 0; t2 < 4; ++t2) {
    float bv = bias1[n0 + t2 * 16 + row];
    if (HASB2) bv += bias2[n0 + t2 * 16 + row];
#pragma unroll
    for (int r = 0; r < 8; ++r) acc[t2][r] = bv;
  }

#pragma unroll 4
  for (int k0 = 0; k0 < K; k0 += 4) {
    const v2f a = *(const v2f*)(ap + k0);
    const v2f b0 = *(const v2f*)(bp0 + k0);
    const v2f b1 = *(const v2f*)(bp0 + btile + k0);
    const v2f b2 = *(const v2f*)(bp0 + 2 * btile + k0);
    const v2f b3 = *(const v2f*)(bp0 + 3 * btile + k0);
    acc[0] = wmma4(a, b0, acc[0]);
    acc[1] = wmma4(a, b1, acc[1]);
    acc[2] = wmma4(a, b2, acc[2]);
    acc[3] = wmma4(a, b3, acc[3]);
  }

#pragma unroll
  for (int t2 = 0; t2 < 4; ++t2) {
    const int col = n0 + t2 * 16 + row;
#pragma unroll
    for (int r = 0; r < 8; ++r) {
      const int orow = m0 + r + half * 8;
      const size_t oi = (size_t)orow * ldo + col;
      float v = acc[t2][r];
      if (HASRES) v += resid[oi];
      out[oi] = v;
    }
  }
}

// ---------------------------------------------------------------------------
// LSTM recurrent scan for one layer, both directions.
// gin: [2 dirs][NSEQ*T_ rows (n*T+t)][256] precomputed x@Wih^T + bih + bhh
// Whh: [2 dirs][256][64]   (gate-major rows)
// hcat: [NSEQ*T_][128]  dir0 -> cols 0..63, dir1 -> cols 64..127
// grid: 22 blocks = 11 n-tiles x 2 dirs; block = 256 threads = 8 waves.
// Per step: g[16,256] = gin_row + h[16,64] @ Whh^T via chained 16x16x4 WMMA.
// ---------------------------------------------------------------------------
__global__ __launch_bounds__(256) void lstm_scan(
    const float* __restrict__ gin,
    const float* __restrict__ Whh,
    float* __restrict__ hcat)
{
  const int d  = blockIdx.x / 11;
  const int n0 = (blockIdx.x % 11) * 16;
  const int wave = threadIdx.x >> 5;
  const int lane = threadIdx.x & 31;
  const int row  = lane & 15;
  const int half = lane >> 4;
  const int koff = half * 2;

  __shared__ float hT[HH_][16];     // h transposed: [hidden][seq-in-tile]
  __shared__ float cS[16][HH_];
  __shared__ float gS[16][G4_];

  for (int i = threadIdx.x; i < 16 * HH_; i += 256) {
    ((float*)hT)[i] = 0.f;
    ((float*)cS)[i] = 0.f;
  }

  const float* gd = gin + (size_t)d * NSEQ * T_ * G4_;
  const float* Wd = Whh + (size_t)d * G4_ * HH_;

  // Preload B fragments (Whh^T) into registers: 2 tiles x 16 k-steps
  v2f rb[2][16];
#pragma unroll
  for (int t2 = 0; t2 < 2; ++t2) {
    const int col = wave * 32 + t2 * 16 + row;   // gate column 0..255
#pragma unroll
    for (int ki = 0; ki < 16; ++ki)
      rb[t2][ki] = *(const v2f*)(Wd + (size_t)col * HH_ + 4 * ki + koff);
  }
  __syncthreads();

  for (int t = 0; t < T_; ++t) {
    const int tt = d ? (T_ - 1 - t) : t;

    v8f acc[2];
#pragma unroll
    for (int t2 = 0; t2 < 2; ++t2) {
      const int col = wave * 32 + t2 * 16 + row;
#pragma unroll
      for (int r = 0; r < 8; ++r) {
        const int nrow = n0 + r + half * 8;
        acc[t2][r] = gd[((size_t)nrow * T_ + tt) * G4_ + col];
      }
    }

#pragma unroll
    for (int ki = 0; ki < 16; ++ki) {
      v2f a;
      a.x = hT[4 * ki + koff][row];
      a.y = hT[4 * ki + koff + 1][row];
      acc[0] = wmma4(a, rb[0][ki], acc[0]);
      acc[1] = wmma4(a, rb[1][ki], acc[1]);
    }

#pragma unroll
    for (int t2 = 0; t2 < 2; ++t2) {
      const int col = wave * 32 + t2 * 16 + row;
#pragma unroll
      for (int r = 0; r < 8; ++r) gS[r + half * 8][col] = acc[t2][r];
    }
    __syncthreads();

    // gate combine: 16 seqs x 64 hidden = 1024 items = 4 per thread
#pragma unroll
    for (int kk = 0; kk < 4; ++kk) {
      const int idx = threadIdx.x + kk * 256;
      const int n   = idx >> 6;
      const int hid = idx & 63;
      const float gi = gS[n][hid];
      const float gf = gS[n][HH_ + hid];
      const float gg = gS[n][2 * HH_ + hid];
      const float go = gS[n][3 * HH_ + hid];
      const float cn = sigm(gf) * cS[n][hid] + sigm(gi) * tanhf(gg);
      const float hn = sigm(go) * tanhf(cn);
      cS[n][hid] = cn;
      hT[hid][n] = hn;
      hcat[(((size_t)(n0 + n)) * T_ + tt) * C_ + d * HH_ + hid] = hn;
    }
    __syncthreads();
  }
}

// ---------------------------------------------------------------------------
// Timestep embedding + per-layer time MLP: te[l][b][c]
// grid = (N_LAYERS * B) blocks of 512 threads
// ---------------------------------------------------------------------------
__global__ __launch_bounds__(512) void tmlp(
    const int* __restrict__ tarr,
    const float* __restrict__ w1, const float* __restrict__ b1,
    const float* __restrict__ w2, const float* __restrict__ b2,
    float* __restrict__ te)
{
  const int l = blockIdx.x >> 1;
  const int b = blockIdx.x & 1;
  __shared__ float te0[C_];
  __shared__ float hid[512];
  const int tid = threadIdx.x;

  if (tid < C_) {
    const int fi = (tid < 64) ? tid : (tid - 64);
    const float fr = __expf(-logf(10000.f) * (float)fi / 64.f);
    const float a = (float)tarr[b] * fr;
    te0[tid] = (tid < 64) ? cosf(a) : sinf(a);
  }
  __syncthreads();

  {
    const float* W1 = w1 + (size_t)l * C_ * 512;
    float s = b1[l * 512 + tid];
    for (int k = 0; k < C_; ++k) s += te0[k] * W1[(size_t)k * 512 + tid];
    hid[tid] = s * sigm(s);   // silu
  }
  __syncthreads();

  if (tid < C_) {
    const float* W2 = w2 + (size_t)l * 512 * C_;
    float s = b2[l * C_ + tid];
    for (int k = 0; k < 512; ++k) s += hid[k] * W2[(size_t)k * C_ + tid];
    te[(size_t)(l * 2 + b) * C_ + tid] = s;
  }
}

// ---------------------------------------------------------------------------
// Layout shuffles / elementwise
// ---------------------------------------------------------------------------
__global__ __launch_bounds__(256) void kTransposeXS(
    const float* __restrict__ x, float* __restrict__ xs)
{
  const int idx = blockIdx.x * 256 + threadIdx.x;     // over B*T*P*C
  if (idx >= B_ * T_ * P_ * C_) return;
  const int c = idx & (C_ - 1);
  const int r = idx >> 7;                 // (b, t, p)
  const int p = r % P_;
  const int t = (r / P_) % T_;
  const int b = r / (P_ * T_);
  xs[(((size_t)(b * P_ + p)) * T_ + t) * C_ + c] = x[idx];
}

__global__ __launch_bounds__(256) void kToGrid(
    const float* __restrict__ hcat, float* __restrict__ hg)
{
  const int idx = blockIdx.x * 256 + threadIdx.x;     // over NSEQ*T*C
  if (idx >= NSEQ * T_ * C_) return;
  const int c = idx & (C_ - 1);
  const int r = idx >> 7;                 // (n, t)
  const int t = r % T_;
  const int n = r / T_;
  const int p = n % P_;
  const int b = n / P_;
  hg[(((size_t)(b * T_ + t)) * P_ + p) * C_ + c] = hcat[idx];
}

__global__ __launch_bounds__(256) void kAddTemb(
    const float* __restrict__ hg, const float* __restrict__ te,
    float* __restrict__ hp)
{
  const int idx = blockIdx.x * 256 + threadIdx.x;     // over NPIX*C
  if (idx >= NPIX * C_) return;
  const int c = idx & (C_ - 1);
  const int b = idx / (T_ * P_ * C_);
  hp[idx] = hg[idx] + te[b * C_ + c];
}

// Transpose a KxN row-major weight into NxK ("Bt") layout.
__global__ __launch_bounds__(256) void kTransW(
    const float* __restrict__ w, float* __restrict__ wt, int K, int N)
{
  const int idx = blockIdx.x * 256 + threadIdx.x;
  if (idx >= K * N) return;
  const int n = idx % N;
  const int k = idx / N;
  wt[(size_t)n * K + k] = w[idx];
}

// ---------------------------------------------------------------------------
// NA2D: one wave per (pixel, head). lane = head-dim channel (DH_=32).
// qkv: [NPIX][384] = q|k|v,  rpb: [NH_][9][9],  out: [NPIX][128]
// ---------------------------------------------------------------------------
__device__ __forceinline__ void axis_nb(int i, int L, int d, int* idx, int* pb) {
  const int g = i % d;
  const int ii = i / d;
  const int Lg = (L - g + d - 1) / d;
  int st = ii - 2;
  if (st < 0) st = 0;
  if (st > Lg - 5) st = Lg - 5;
#pragma unroll
  for (int u = 0; u < 5; ++u) {
    idx[u] = g + d * (st + u);
    pb[u] = 4 - (ii - st) + u;
  }
}

__global__ __launch_bounds__(256) void natten(
    const float* __restrict__ qkv, const float* __restrict__ rpb,
    float* __restrict__ outp, int dil)
{
  const int wave = threadIdx.x >> 5;
  const int lane = threadIdx.x & 31;
  const int gw = blockIdx.x * 8 + wave;
  const int head = gw & 3;
  const int pix = gw >> 2;              // (b*H + i)*W + j
  const int j = pix % P_;
  const int bi = pix / P_;
  const int i = bi % T_;
  const int b = bi / T_;

  int ih[5], pbh[5], iw[5], pbw[5];
  axis_nb(i, T_, dil, ih, pbh);
  axis_nb(j, P_, dil, iw, pbw);

  const float scale = 0.17677669529663687f;   // 32^-0.5
  const float q = qkv[(size_t)pix * 384 + head * DH_ + lane] * scale;
  const float* rp = rpb + head * 81;

  float sc[25];
#pragma unroll
  for (int a = 0; a < 5; ++a)
#pragma unroll
    for (int bb = 0; bb < 5; ++bb) {
      const int npix = (b * T_ + ih[a]) * P_ + iw[bb];
      const float kv = qkv[(size_t)npix * 384 + C_ + head * DH_ + lane];
      sc[a * 5 + bb] = waveSum(q * kv) + rp[pbh[a] * 9 + pbw[bb]];
    }

  float mx = sc[0];
#pragma unroll
  for (int u = 1; u < 25; ++u) mx = fmaxf(mx, sc[u]);
  float sum = 0.f;
#pragma unroll
  for (int u = 0; u < 25; ++u) { sc[u] = __expf(sc[u] - mx); sum += sc[u]; }
  const float inv = 1.f / sum;

  float o = 0.f;
#pragma unroll
  for (int a = 0; a < 5; ++a)
#pragma unroll
    for (int bb = 0; bb < 5; ++bb) {
      const int npix = (b * T_ + ih[a]) * P_ + iw[bb];
      o += sc[a * 5 + bb] * qkv[(size_t)npix * 384 + 2 * C_ + head * DH_ + lane];
    }
  outp[(size_t)pix * C_ + head * DH_ + lane] = o * inv;
}

// ---------------------------------------------------------------------------
// Host-side orchestration
// ---------------------------------------------------------------------------
extern "C" void kernel_launch(void* const* d_in, const int* in_sizes, int n_in,
                              void* d_out, int out_size, void* d_ws, size_t ws_size,
                              hipStream_t stream) {
  (void)in_sizes; (void)n_in; (void)out_size; (void)ws_size;

  const float* x    = (const float*)d_in[0];
  const int*   tarr = (const int*)  d_in[1];
  const float* Wih  = (const float*)d_in[2];   // (2,2,256,128)  [n][k] already
  const float* Whh  = (const float*)d_in[3];   // (2,2,256,64)
  const float* bih  = (const float*)d_in[4];   // (2,2,256)
  const float* bhh  = (const float*)d_in[5];   // (2,2,256)
  const float* wqkv = (const float*)d_in[6];   // (2,128,384)
  const float* bqkv = (const float*)d_in[7];   // (2,384)
  const float* wprj = (const float*)d_in[8];   // (2,128,128)
  const float* bprj = (const float*)d_in[9];   // (2,128)
  const float* rpb  = (const float*)d_in[10];  // (2,4,9,9)
  const float* tw1  = (const float*)d_in[11];  // (2,128,512)
  const float* tb1  = (const float*)d_in[12];  // (2,512)
  const float* tw2  = (const float*)d_in[13];  // (2,512,128)
  const float* tb2  = (const float*)d_in[14];  // (2,128)
  float* out = (float*)d_out;
  float* ws  = (float*)d_ws;

  // Workspace arena (floats). xs is reused as Hgrid; gin reused as qkv.
  const size_t SZ    = (size_t)NPIX * C_;      // 2,883,584
  const size_t o_xs  = 0;                      // xs -> later Hgrid
  const size_t o_hA  = SZ;                     // hcat layer 0
  const size_t o_hB  = 2 * SZ;                 // hcat layer 1
  const size_t o_gin = 3 * SZ;                 // 2 x NPIX*256 -> later qkv
  const size_t gin_d = (size_t)NPIX * G4_;     // 5,767,168
  const size_t o_hp  = o_gin + 2 * gin_d;      // h + temb
  const size_t o_at  = o_hp + SZ;              // attention output
  const size_t o_te  = o_at + SZ;              // 2 layers x B x C = 512
  const size_t o_wqT = o_te + 512;             // 2 x 384 x 128 = 98304
  const size_t o_wpT = o_wqT + 2 * 384 * C_;   // 2 x 128 x 128 = 32768

  const int ELGRID = (NPIX * C_) / 256;        // 11264, exact

  // Transpose attention weights into [n][k] once (tiny)
  for (int l = 0; l < 2; ++l) {
    kTransW<<<(C_ * 384) / 256, 256, 0, stream>>>(
        wqkv + (size_t)l * C_ * 384, ws + o_wqT + (size_t)l * 384 * C_, C_, 384);
    kTransW<<<(C_ * C_) / 256, 256, 0, stream>>>(
        wprj + (size_t)l * C_ * C_, ws + o_wpT + (size_t)l * C_ * C_, C_, C_);
  }

  // Time-embedding MLPs for both layers
  tmlp<<<4, 512, 0, stream>>>(tarr, tw1, tb1, tw2, tb2, ws + o_te);
  // (B,T*P,C) -> (B*P, T, C) sequences
  kTransposeXS<<<ELGRID, 256, 0, stream>>>(x, ws + o_xs);

  // ---- BiLSTM: 2 layers ----
  for (int l = 0; l < 2; ++l) {
    const float* Ain = (l == 0) ? (ws + o_xs) : (ws + o_hA);
    float* hout = (l == 0) ? (ws + o_hA) : (ws + o_hB);
    for (int d = 0; d < 2; ++d) {
      // gin = Ain @ Wih^T + bih + bhh   (M=22528,N=256,K=128)
      gemm16t<true, false><<<704, 256, 0, stream>>>(
          Ain, C_,
          Wih + (size_t)(l * 2 + d) * G4_ * C_, C_,
          bih + (size_t)(l * 2 + d) * G4_, bhh + (size_t)(l * 2 + d) * G4_,
          nullptr, ws + o_gin + (size_t)d * gin_d, G4_, /*NG=*/4, /*K=*/C_);
    }
    lstm_scan<<<22, 256, 0, stream>>>(
        ws + o_gin, Whh + (size_t)l * 2 * G4_ * HH_, hout);
  }

  // (B*P, T, C) -> (B, T, P, C) grid (aliases dead xs buffer)
  kToGrid<<<ELGRID, 256, 0, stream>>>(ws + o_hB, ws + o_xs);
  float* Hg   = ws + o_xs;
  float* qkvB = ws + o_gin;

  // ---- 2 NA2D layers ----
  for (int l = 0; l < 2; ++l) {
    kAddTemb<<<ELGRID, 256, 0, stream>>>(Hg, ws + o_te + (size_t)l * 2 * C_,
                                         ws + o_hp);
    // qkv = (h+temb) @ wqkv + bqkv   (M=22528,N=384,K=128)
    gemm16t<false, false><<<1056, 256, 0, stream>>>(
        ws + o_hp, C_,
        ws + o_wqT + (size_t)l * 384 * C_, C_,
        bqkv + (size_t)l * 384, nullptr, nullptr, qkvB, 384, /*NG=*/6, C_);
    natten<<<ELGRID, 256, 0, stream>>>(qkvB, rpb + (size_t)l * NH_ * 81,
                                       ws + o_at, (l == 0) ? 1 : 2);
    // h = h + attnout @ wproj + bproj   (M=22528,N=128,K=128)
    gemm16t<false, true><<<352, 256, 0, stream>>>(
        ws + o_at, C_,
        ws + o_wpT + (size_t)l * C_ * C_, C_,
        bprj + (size_t)l * C_, nullptr, /*resid=*/Hg,
        (l == 0) ? Hg : out, C_, /*NG=*/2, C_);
  }
}